// SPDDecoder_28140625723423
// MI455X (gfx1250) — compile-verified
//
#include <hip/hip_runtime.h>
#include <hip/hip_bf16.h>
#include <math.h>

typedef __attribute__((ext_vector_type(2))) float v2f;
typedef __attribute__((ext_vector_type(8))) float v8f;

#define EPS_REEIG 1e-4f

static __device__ __forceinline__ int imin_i(int a, int b) { return a < b ? a : b; }

// ---------------- Jacobi eigendecomposition (branchless, fixed sweeps) ----------
template<int N>
static __device__ __forceinline__ void jacobi_eig(float (&A)[N][N], float (&V)[N][N], int sweeps) {
#pragma unroll
  for (int i = 0; i < N; ++i)
#pragma unroll
    for (int j = 0; j < N; ++j)
      V[i][j] = (i == j) ? 1.0f : 0.0f;

  for (int s = 0; s < sweeps; ++s) {
#pragma unroll
    for (int p = 0; p < N - 1; ++p) {
#pragma unroll
      for (int q = p + 1; q < N; ++q) {
        float apq = A[p][q];
        float d2  = 0.5f * (A[q][q] - A[p][p]);
        float hyp = sqrtf(d2 * d2 + apq * apq);
        float sgn = (d2 >= 0.0f) ? 1.0f : -1.0f;
        float t   = (apq * sgn) / (fabsf(d2) + hyp + 1e-30f);
        float c   = 1.0f / sqrtf(1.0f + t * t);
        float sn  = t * c;
        // A <- A * J   (columns p,q)
#pragma unroll
        for (int k = 0; k < N; ++k) {
          float akp = A[k][p], akq = A[k][q];
          A[k][p] = c * akp - sn * akq;
          A[k][q] = sn * akp + c * akq;
        }
        // A <- J^T * A (rows p,q)
#pragma unroll
        for (int k = 0; k < N; ++k) {
          float apk = A[p][k], aqk = A[q][k];
          A[p][k] = c * apk - sn * aqk;
          A[q][k] = sn * apk + c * aqk;
        }
        // V <- V * J
#pragma unroll
        for (int k = 0; k < N; ++k) {
          float vkp = V[k][p], vkq = V[k][q];
          V[k][p] = c * vkp - sn * vkq;
          V[k][q] = sn * vkp + c * vkq;
        }
      }
    }
  }
}

// vech (row-major upper triangle) <-> full symmetric
template<int N, int NV>
static __device__ __forceinline__ void vech_to_full(const float (&v)[NV], float (&A)[N][N]) {
  int idx = 0;
#pragma unroll
  for (int i = 0; i < N; ++i)
#pragma unroll
    for (int j = i; j < N; ++j) {
      A[i][j] = v[idx];
      A[j][i] = v[idx];
      ++idx;
    }
}

// out = vech( V * diag(fw) * V^T )
template<int N, int NV>
static __device__ __forceinline__ void reconstruct_vech(const float (&V)[N][N], const float (&fw)[N],
                                                        float (&out)[NV]) {
  int idx = 0;
#pragma unroll
  for (int i = 0; i < N; ++i)
#pragma unroll
    for (int j = i; j < N; ++j) {
      float s = 0.0f;
#pragma unroll
      for (int k = 0; k < N; ++k) s = fmaf(fw[k] * V[i][k], V[j][k], s);
      out[idx++] = s;
    }
}

// ---------------- Build vech-space bimap matrix M from W [n_out, n_in] --------
// vech(W X W^T)[ (i,j) ] = sum_{a<=b} M[(i,j)][(a,b)] * vech(X)[(a,b)]
static __device__ void build_M(const float* __restrict__ W, int n_out, int n_in,
                               float* __restrict__ Mld, int tid, int nthreads) {
  const int MV = n_out * (n_out + 1) / 2;
  const int KV = n_in * (n_in + 1) / 2;
  for (int idx = tid; idx < MV * KV; idx += nthreads) {
    int row = idx / KV, col = idx % KV;
    int i = 0, r = row;
    while (r >= n_out - i) { r -= (n_out - i); ++i; }
    int j = i + r;
    int a = 0, cc = col;
    while (cc >= n_in - a) { cc -= (n_in - a); ++a; }
    int b = a + cc;
    float m = W[i * n_in + a] * W[j * n_in + b];
    if (a != b) m += W[i * n_in + b] * W[j * n_in + a];
    Mld[idx] = m;
  }
}

// ---------------- WMMA vech-space bimap: vout[MV] = M[MV,KV] * vin[KV] ---------
// Per-wave: 32 items (one per lane). Two 16-item groups (N dim), RT row tiles
// (M dim), KC chunks of K=4 along KV. Operands staged via LDS to match the
// 16x16x4 f32 WMMA lane layouts (A: lane&15 = M row, VGPR r = K r/r+2 per
// half-wave; B: same K pattern, lane&15 = N col; C: VGPR r = rows r / r+8).
template<int KV, int MV, int KC, int RT>
static __device__ __forceinline__ void bimap_wmma(const float* __restrict__ Mld,
                                                  float* __restrict__ inW,   // wave region, 32*KV used
                                                  float* __restrict__ tileW, // wave region, 256 floats
                                                  const float (&vin)[KV], float (&vout)[MV], int lane) {
#pragma unroll
  for (int k = 0; k < KV; ++k) inW[lane * KV + k] = vin[k];
  __syncthreads();

#if __has_builtin(__builtin_amdgcn_wmma_f32_16x16x4_f32)
  const int half = lane >> 4;
  const int nloc = lane & 15;
#pragma unroll
  for (int rt = 0; rt < RT; ++rt) {
    for (int g = 0; g < 2; ++g) {       // item group: items g*16 .. g*16+15
      v8f acc = {};
      for (int ck = 0; ck < KC; ++ck) { // K chunks of 4
        v2f av, bv;
#pragma unroll
        for (int r = 0; r < 2; ++r) {
          int kk    = ck * 4 + r + 2 * half;
          int ksafe = imin_i(kk, KV - 1);
          int rowA  = rt * 16 + nloc;
          float a = Mld[imin_i(rowA, MV - 1) * KV + ksafe];
          av[r] = (rowA < MV && kk < KV) ? a : 0.0f;
          float b = inW[(g * 16 + nloc) * KV + ksafe];
          bv[r] = (kk < KV) ? b : 0.0f;
        }
        // D = A(16x4) * B(4x16) + C, full fp32
        acc = __builtin_amdgcn_wmma_f32_16x16x4_f32(false, av, false, bv,
                                                    (short)0, acc, false, false);
      }
      // C tile -> LDS (VGPR r holds rows r (lanes 0-15) and r+8 (lanes 16-31))
#pragma unroll
      for (int r = 0; r < 8; ++r) tileW[(r + 8 * half) * 16 + nloc] = acc[r];
      __syncthreads();
      // redistribute: lane l owns item l; its columns live at nloc in group (l>>4)
#pragma unroll
      for (int r2 = 0; r2 < 16; ++r2) {
        int R = rt * 16 + r2;            // compile-time after unroll
        if (R < MV) {
          float v = tileW[r2 * 16 + nloc];
          if (half == g) vout[R] = v;
        }
      }
      __syncthreads();
    }
  }
#else
  // Fallback scalar matvec (keeps compile green if builtin signature differs)
#pragma unroll
  for (int m = 0; m < MV; ++m) {
    float s = 0.0f;
    for (int k = 0; k < KV; ++k) s = fmaf(Mld[m * KV + k], vin[k], s);
    vout[m] = s;
  }
  __syncthreads();
  __syncthreads();
  // keep barrier counts irrelevant: fallback path is compile-time uniform anyway
#endif
}

// ---------------- Kernel --------------------------------------------------------
__global__ void __launch_bounds__(256)
spdnet_wmma_kernel(const float* __restrict__ vech, const float* __restrict__ W1,
                   const float* __restrict__ W2, const float* __restrict__ W3,
                   float* __restrict__ out, int B) {
  __shared__ float sM1[15 * 6];
  __shared__ float sM2[28 * 15];
  __shared__ float sM3[45 * 28];
  __shared__ float sIn[8 * 32 * 28];   // per-wave input staging (max KV = 28)
  __shared__ float sTile[8 * 256];     // per-wave 16x16 C tile

  const int tidLocal = threadIdx.x;
  build_M(W1, 5, 3, sM1, tidLocal, 256);
  build_M(W2, 7, 5, sM2, tidLocal, 256);
  build_M(W3, 9, 7, sM3, tidLocal, 256);
  __syncthreads();

  const int tid  = blockIdx.x * 256 + tidLocal;
  const int lane = tidLocal & 31;
  const int wid  = tidLocal >> 5;
  float* inW   = &sIn[wid * 32 * 28];
  float* tileW = &sTile[wid * 256];

  const int t = imin_i(tid, B - 1);
  const float* vp = vech + (size_t)t * 6;
  float v6[6];
#pragma unroll
  for (int k = 0; k < 6; ++k) v6[k] = vp[k];

  // ---- ExpEig on 3x3 ----
  float A3[3][3], V3[3][3];
  vech_to_full<3, 6>(v6, A3);
  jacobi_eig<3>(A3, V3, 5);
  float fw3[3];
#pragma unroll
  for (int i = 0; i < 3; ++i) fw3[i] = expf(A3[i][i]);
  float x3v[6];
  reconstruct_vech<3, 6>(V3, fw3, x3v);

  // ---- BiMap 1 (WMMA): vech6 -> vech15 ----
  float x5v[15];
  bimap_wmma<6, 15, 2, 1>(sM1, inW, tileW, x3v, x5v, lane);

  // ---- ReEig on 5x5 ----
  float A5[5][5], V5[5][5];
  vech_to_full<5, 15>(x5v, A5);
  jacobi_eig<5>(A5, V5, 6);
  float fw5[5];
#pragma unroll
  for (int i = 0; i < 5; ++i) fw5[i] = fmaxf(A5[i][i], EPS_REEIG);
  float y5v[15];
  reconstruct_vech<5, 15>(V5, fw5, y5v);

  // ---- BiMap 2 (WMMA): vech15 -> vech28 ----
  float x7v[28];
  bimap_wmma<15, 28, 4, 2>(sM2, inW, tileW, y5v, x7v, lane);

  // ---- ReEig on 7x7 ----
  float A7[7][7], V7[7][7];
  vech_to_full<7, 28>(x7v, A7);
  jacobi_eig<7>(A7, V7, 7);
  float fw7[7];
#pragma unroll
  for (int i = 0; i < 7; ++i) fw7[i] = fmaxf(A7[i][i], EPS_REEIG);
  float y7v[28];
  reconstruct_vech<7, 28>(V7, fw7, y7v);

  // ---- BiMap 3 (WMMA): vech28 -> vech45 ----
  float x9v[45];
  bimap_wmma<28, 45, 7, 3>(sM3, inW, tileW, y7v, x9v, lane);

  // ---- ReEig on 9x9, write full matrix ----
  float A9[9][9], V9[9][9];
  vech_to_full<9, 45>(x9v, A9);
  jacobi_eig<9>(A9, V9, 8);
  float fw9[9];
#pragma unroll
  for (int i = 0; i < 9; ++i) fw9[i] = fmaxf(A9[i][i], EPS_REEIG);

  if (tid < B) {
    float* op = out + (size_t)tid * 81;
#pragma unroll
    for (int i = 0; i < 9; ++i)
#pragma unroll
      for (int j = 0; j < 9; ++j) {
        float s = 0.0f;
#pragma unroll
        for (int k = 0; k < 9; ++k) s = fmaf(fw9[k] * V9[i][k], V9[j][k], s);
        op[i * 9 + j] = s;
      }
  }
}

// ---------------- Launcher ------------------------------------------------------
extern "C" void kernel_launch(void* const* d_in, const int* in_sizes, int n_in,
                              void* d_out, int out_size, void* d_ws, size_t ws_size,
                              hipStream_t stream) {
  (void)n_in; (void)out_size; (void)d_ws; (void)ws_size;
  const float* vech = (const float*)d_in[0];
  const float* W1   = (const float*)d_in[1];
  const float* W2   = (const float*)d_in[2];
  const float* W3   = (const float*)d_in[3];
  float* out = (float*)d_out;
  const int B = in_sizes[0] / 6;
  const int grid = (B + 255) / 256;
  hipLaunchKernelGGL(spdnet_wmma_kernel, dim3(grid), dim3(256), 0, stream,
                     vech, W1, W2, W3, out, B);
}